// GCNModel_25795573580199
// MI455X (gfx1250) — compile-verified
//
#include <hip/hip_runtime.h>
#include <math.h>

typedef __attribute__((ext_vector_type(2))) float v2f;
typedef __attribute__((ext_vector_type(8))) float v8f;

#define N_NODES 100000
#define N_EDGES 1200000

// ---------------- utility kernels ----------------

__global__ void k_fill(float* __restrict__ p, float v, int n) {
    int i = blockIdx.x * blockDim.x + threadIdx.x;
    if (i < n) p[i] = v;
}

__global__ void k_deg(const long long* __restrict__ dst, float* __restrict__ deg, int E) {
    int e = blockIdx.x * blockDim.x + threadIdx.x;
    if (e < E) atomicAdd(&deg[dst[e]], 1.0f);
}

__global__ void k_rsqrt(float* __restrict__ p, int n) {
    int i = blockIdx.x * blockDim.x + threadIdx.x;
    if (i < n) p[i] = rsqrtf(p[i]);
}

__global__ void k_edge_norm(const long long* __restrict__ src, const long long* __restrict__ dst,
                            const float* __restrict__ dis, float* __restrict__ norm, int E) {
    int e = blockIdx.x * blockDim.x + threadIdx.x;
    if (e < E) norm[e] = dis[src[e]] * dis[dst[e]];
}

// ---------------- fp32 WMMA GEMM: C[M x F] = A[M x K] * B[K x F] ----------------
// K, F compile-time so the K-loop fully unrolls: per step one b64 A-load and two
// b32 B-loads with immediate offsets, then V_WMMA_F32_16X16X4_F32.
// Guarding is done by CLAMPING indices (row m of A only feeds row m of C; col n
// of B only feeds col n of C; clamped rows/cols are never stored), so the hot
// loop has no EXEC manipulation and no branches.
// blockDim.x = 32 * ceil(F/16); blockIdx.x = M tile.
// A-frag (16x4 f32): lane L -> row L%16, K pair {2*(L/16), 2*(L/16)+1}.
// B-frag (4x16 f32): lane L -> col L%16, K pair {2*(L/16), 2*(L/16)+1}.
// C/D (16x16 f32, 8 VGPRs): VGPR i -> row 8*(L/16)+i, col L%16.
template <int K, int F>
__global__ void k_gemm_wmma(const float* __restrict__ A, const float* __restrict__ B,
                            float* __restrict__ C, int M) {
    const int lane  = threadIdx.x & 31;
    const int ntile = threadIdx.x >> 5;
    const int mtile = blockIdx.x;

    const int mloc = lane & 15;
    const int half = lane >> 4;      // 0 or 1
    const int koff = half * 2;       // K pair base within each K-step of 4

    int m = mtile * 16 + mloc;
    if (m >= M) m = M - 1;                       // clamp (never stored)
    int n = ntile * 16 + mloc;
    const int nc = (n < F) ? n : (F - 1);        // clamp (never stored)

    const float* __restrict__ arow = A + (size_t)m * K + koff;
    const float* __restrict__ bcol = B + (size_t)koff * F + nc;

    v8f acc = {};
    #pragma unroll
    for (int kk = 0; kk < K; kk += 4) {
        v2f a = *(const v2f*)(arow + kk);        // 8B contiguous, immediate offset
        v2f b;
        b.x = bcol[(size_t)kk * F];              // immediate offsets, clause-able
        b.y = bcol[(size_t)(kk + 1) * F];
        acc = __builtin_amdgcn_wmma_f32_16x16x4_f32(
            /*neg_a=*/false, a, /*neg_b=*/false, b,
            /*c_mod=*/(short)0, acc, /*reuse_a=*/false, /*reuse_b=*/false);
    }

    if (n < F) {
        #pragma unroll
        for (int i = 0; i < 8; ++i) {
            int row = mtile * 16 + half * 8 + i;
            if (row < M) C[(size_t)row * F + n] = acc[i];
        }
    }
}

// ---------------- aggregation ----------------

// out[i,f] = h[i,f] * dis[i]^2 + bias[f]   (self-loop contribution + bias; also
// fully initializes the output buffer for the subsequent atomic scatter)
__global__ void k_self_init(const float* __restrict__ h, const float* __restrict__ dis,
                            const float* __restrict__ bias, float* __restrict__ out,
                            int Nn, int F) {
    int i = blockIdx.x * blockDim.x + threadIdx.x;
    int total = Nn * F;
    if (i >= total) return;
    int node = i / F;
    int f = i - node * F;
    float d = dis[node];
    out[i] = h[i] * d * d + bias[f];
}

// per (edge, 4-feature chunk): out[dst] += h[src] * norm[e]
// Consecutive threads cover consecutive chunks of one edge -> the per-edge
// src/dst/norm loads coalesce to one request, gathers are 16B contiguous,
// atomics land in the 192MB L2 (feature buffers are only 25.6MB).
__global__ void k_scatter(const float* __restrict__ h, const long long* __restrict__ src,
                          const long long* __restrict__ dst, const float* __restrict__ norm,
                          float* __restrict__ out, int E, int F4, int F) {
    int gid = blockIdx.x * blockDim.x + threadIdx.x;
    int total = E * F4;
    if (gid >= total) return;
    int e = gid / F4;
    int c = (gid - e * F4) * 4;
    long long s = src[e];
    long long d = dst[e];
    float nv = norm[e];
    const float4 hv = *(const float4*)(h + (size_t)s * F + c);
    float* o = out + (size_t)d * F + c;
    atomicAdd(o + 0, hv.x * nv);
    atomicAdd(o + 1, hv.y * nv);
    atomicAdd(o + 2, hv.z * nv);
    atomicAdd(o + 3, hv.w * nv);
}

__global__ void k_relu(float* __restrict__ p, int n) {
    int i = blockIdx.x * blockDim.x + threadIdx.x;
    if (i < n) p[i] = fmaxf(p[i], 0.0f);
}

// ---------------- log-softmax: one wave32 per node, F=40 classes ----------------
__global__ void k_logsoftmax(float* __restrict__ out, int Nn, int F) {
    int gid  = blockIdx.x * blockDim.x + threadIdx.x;
    int wave = gid >> 5;
    int lane = threadIdx.x & 31;
    if (wave >= Nn) return;
    float* row = out + (size_t)wave * F;

    float v0 = (lane < F)      ? row[lane]      : -__builtin_inff();
    float v1 = (lane + 32 < F) ? row[lane + 32] : -__builtin_inff();

    float m = fmaxf(v0, v1);
    #pragma unroll
    for (int o = 16; o > 0; o >>= 1) m = fmaxf(m, __shfl_xor(m, o, 32));

    float s = ((lane < F) ? expf(v0 - m) : 0.0f) +
              ((lane + 32 < F) ? expf(v1 - m) : 0.0f);
    #pragma unroll
    for (int o = 16; o > 0; o >>= 1) s += __shfl_xor(s, o, 32);

    float lse = m + logf(s);
    if (lane < F)      row[lane]      = v0 - lse;
    if (lane + 32 < F) row[lane + 32] = v1 - lse;
}

// ---------------- host launcher ----------------

extern "C" void kernel_launch(void* const* d_in, const int* in_sizes, int n_in,
                              void* d_out, int out_size, void* d_ws, size_t ws_size,
                              hipStream_t stream) {
    (void)in_sizes; (void)n_in; (void)out_size; (void)ws_size;

    const float*     x   = (const float*)d_in[0];
    const long long* ei  = (const long long*)d_in[1];   // int64 [2, E]
    const float*     W1  = (const float*)d_in[2];
    const float*     b1  = (const float*)d_in[3];
    const float*     W2  = (const float*)d_in[4];
    const float*     b2  = (const float*)d_in[5];
    const float*     W3  = (const float*)d_in[6];
    const float*     b3  = (const float*)d_in[7];
    float*           out = (float*)d_out;

    const int Nn = N_NODES, E = N_EDGES;
    const long long* src = ei;
    const long long* dst = ei + E;

    // workspace carve-out (256B aligned): dis | norm | G (gemm out) | A (agg out)
    char*  ws  = (char*)d_ws;
    size_t off = 0;
    auto carve = [&](size_t bytes) -> float* {
        float* p = (float*)(ws + off);
        off += (bytes + 255) & ~(size_t)255;
        return p;
    };
    float* dis  = carve((size_t)Nn * 4);
    float* norm = carve((size_t)E * 4);
    float* G    = carve((size_t)Nn * 64 * 4);
    float* Abuf = carve((size_t)Nn * 64 * 4);

    const int T = 256;
    const int mtiles = (Nn + 15) / 16;

    // degrees + symmetric norm
    k_fill     <<<(Nn + T - 1) / T, T, 0, stream>>>(dis, 1.0f, Nn);   // self-loop deg
    k_deg      <<<(E + T - 1) / T, T, 0, stream>>>(dst, dis, E);
    k_rsqrt    <<<(Nn + T - 1) / T, T, 0, stream>>>(dis, Nn);
    k_edge_norm<<<(E + T - 1) / T, T, 0, stream>>>(src, dst, dis, norm, E);

    // ---- layer 1: relu(agg(x @ W1) + b1), F: 128 -> 64 ----
    k_gemm_wmma<128, 64><<<mtiles, 4 * 32, 0, stream>>>(x, W1, G, Nn);
    k_self_init<<<(Nn * 64 + T - 1) / T, T, 0, stream>>>(G, dis, b1, Abuf, Nn, 64);
    k_scatter  <<<(E * 16 + T - 1) / T, T, 0, stream>>>(G, src, dst, norm, Abuf, E, 16, 64);
    k_relu     <<<(Nn * 64 + T - 1) / T, T, 0, stream>>>(Abuf, Nn * 64);

    // ---- layer 2: relu(agg(h @ W2) + b2), F: 64 -> 64 ----
    k_gemm_wmma<64, 64><<<mtiles, 4 * 32, 0, stream>>>(Abuf, W2, G, Nn);
    k_self_init<<<(Nn * 64 + T - 1) / T, T, 0, stream>>>(G, dis, b2, Abuf, Nn, 64);
    k_scatter  <<<(E * 16 + T - 1) / T, T, 0, stream>>>(G, src, dst, norm, Abuf, E, 16, 64);
    k_relu     <<<(Nn * 64 + T - 1) / T, T, 0, stream>>>(Abuf, Nn * 64);

    // ---- layer 3: agg(h @ W3) + b3, F: 64 -> 40, into d_out ----
    k_gemm_wmma<64, 40><<<mtiles, 3 * 32, 0, stream>>>(Abuf, W3, G, Nn);
    k_self_init<<<(Nn * 40 + T - 1) / T, T, 0, stream>>>(G, dis, b3, out, Nn, 40);
    k_scatter  <<<(E * 10 + T - 1) / T, T, 0, stream>>>(G, src, dst, norm, out, E, 10, 40);

    // log-softmax over 40 classes, one wave32 per node
    k_logsoftmax<<<(Nn * 32 + T - 1) / T, T, 0, stream>>>(out, Nn, 40);
}